// VanillaRNN_8375186227537
// MI455X (gfx1250) — compile-verified
//
#include <hip/hip_runtime.h>

typedef float v2f __attribute__((ext_vector_type(2)));
typedef float v8f __attribute__((ext_vector_type(8)));

#define B_ 2048
#define S_ 125
#define I_ 2
#define H_ 300
#define O_ 2

#define BT        32      // batch rows per workgroup
#define NTILES    19      // ceil(H/16)
#define HPAD      308     // LDS row stride in floats (308 mod 64 = 52 -> conflict-friendly)
#define NTHREADS  256
#define NWAVES    8

__global__ __launch_bounds__(NTHREADS)
void rnn_fused_kernel(const float* __restrict__ x,      // [B,S,I]
                      const float* __restrict__ noise,  // [B,S,H]
                      const float* __restrict__ W_ih,   // [H,I]
                      const float* __restrict__ b_ih,   // [H]
                      const float* __restrict__ W_hh,   // [H,H]
                      const float* __restrict__ b_hh,   // [H]
                      const float* __restrict__ W_out,  // [O,H]
                      const float* __restrict__ b_out,  // [O]
                      float* __restrict__ out)          // [B,S,O]
{
    __shared__ float hbuf[2][BT][HPAD];

    const int tid  = threadIdx.x;
    const int wave = tid >> 5;
    const int lane = tid & 31;
    const int lo16 = lane & 15;
    const int hi   = lane >> 4;          // 0 or 1 (wave half)
    const int b0   = blockIdx.x * BT;

    // zero-init both h buffers (h0 = 0, padding columns = 0)
    for (int i = tid; i < 2 * BT * HPAD; i += NTHREADS)
        ((float*)hbuf)[i] = 0.0f;

    // preload W_out columns for the per-step output reduction
    float wo0[10], wo1[10];
#pragma unroll
    for (int j = 0; j < 10; ++j) {
        int n = lane + 32 * j;
        wo0[j] = (n < H_) ? W_out[n]      : 0.0f;
        wo1[j] = (n < H_) ? W_out[H_ + n] : 0.0f;
    }
    const float bo0 = b_out[0];
    const float bo1 = b_out[1];

    __syncthreads();

    int cur = 0;
    for (int s = 0; s < S_; ++s) {
        const int nxt = cur ^ 1;

        // ---------- recurrence: h_new = relu(h @ W_hh^T + xin + b_hh) + 0.01*noise ----------
        for (int nt = wave; nt < NTILES; nt += NWAVES) {
            const int  n    = nt * 16 + lo16;          // D column handled by this lane
            const bool nok  = (n < H_);
            // clamp row pointer for padded columns: garbage lands only in
            // D columns n>=300, which are never stored.
            const float* wrow  = W_hh + (size_t)(nok ? n : (H_ - 1)) * H_;
            const float* arow0 = &hbuf[cur][lo16][0];        // M-tile 0 (rows 0..15)
            const float* arow1 = &hbuf[cur][16 + lo16][0];   // M-tile 1 (rows 16..31)
            const int    ksh   = hi * 2;

            v8f acc0 = {0.f,0.f,0.f,0.f,0.f,0.f,0.f,0.f};
            v8f acc1 = {0.f,0.f,0.f,0.f,0.f,0.f,0.f,0.f};

#pragma unroll 5
            for (int k0 = 0; k0 < H_; k0 += 4) {
                const int kk = k0 + ksh;
                // B-fragment: B(k,n) = W_hh[n][k] ; lane holds K = {kk, kk+1}
                v2f bfrag = *(const v2f*)(wrow + kk);
                // A-fragments from LDS: lane holds h[row][kk], h[row][kk+1]
                v2f a0 = *(const v2f*)(arow0 + kk);
                v2f a1 = *(const v2f*)(arow1 + kk);
                acc0 = __builtin_amdgcn_wmma_f32_16x16x4_f32(
                           false, a0, false, bfrag, (short)0, acc0, false, false);
                acc1 = __builtin_amdgcn_wmma_f32_16x16x4_f32(
                           false, a1, false, bfrag, (short)0, acc1, false, false);
            }

            // ---------- epilogue: bias + xin + relu + noise, write h_new ----------
            if (nok) {
                const float wih0 = W_ih[n * 2 + 0];
                const float wih1 = W_ih[n * 2 + 1];
                const float bih  = b_ih[n];
                const float bhh  = b_hh[n];
#pragma unroll
                for (int r = 0; r < 8; ++r) {
                    const int m = r + 8 * hi;            // D row within 16x16 tile
                    // M-tile 0
                    {
                        const int    b   = b0 + m;
                        const v2f    xv  = *(const v2f*)(x + ((size_t)b * S_ + s) * I_);
                        const float  xin = xv.x * wih0 + xv.y * wih1 + bih;
                        float hv = acc0[r] + xin + bhh;
                        hv = fmaxf(hv, 0.0f);
                        hv += 0.01f * noise[((size_t)b * S_ + s) * H_ + n];
                        hbuf[nxt][m][n] = hv;
                    }
                    // M-tile 1
                    {
                        const int    row = 16 + m;
                        const int    b   = b0 + row;
                        const v2f    xv  = *(const v2f*)(x + ((size_t)b * S_ + s) * I_);
                        const float  xin = xv.x * wih0 + xv.y * wih1 + bih;
                        float hv = acc1[r] + xin + bhh;
                        hv = fmaxf(hv, 0.0f);
                        hv += 0.01f * noise[((size_t)b * S_ + s) * H_ + n];
                        hbuf[nxt][row][n] = hv;
                    }
                }
            }
        }

        __syncthreads();   // h_new fully written; one barrier per step suffices

        // ---------- per-step output: out[b,s,:] = h_new[b,:] @ W_out^T + b_out ----------
#pragma unroll
        for (int q = 0; q < 4; ++q) {
            const int row = wave * 4 + q;     // 8 waves x 4 rows = 32 rows
            float p0 = 0.0f, p1 = 0.0f;
#pragma unroll
            for (int j = 0; j < 10; ++j) {
                const int n = lane + 32 * j;
                const float hv = (n < H_) ? hbuf[nxt][row][n] : 0.0f;
                p0 += hv * wo0[j];
                p1 += hv * wo1[j];
            }
#pragma unroll
            for (int off = 16; off > 0; off >>= 1) {
                p0 += __shfl_xor(p0, off, 32);
                p1 += __shfl_xor(p1, off, 32);
            }
            if (lane == 0) {
                const int    b    = b0 + row;
                const size_t oidx = ((size_t)b * S_ + s) * O_;
                out[oidx + 0] = p0 + bo0;
                out[oidx + 1] = p1 + bo1;
            }
        }

        cur = nxt;
    }
}

extern "C" void kernel_launch(void* const* d_in, const int* in_sizes, int n_in,
                              void* d_out, int out_size, void* d_ws, size_t ws_size,
                              hipStream_t stream) {
    (void)in_sizes; (void)n_in; (void)d_ws; (void)ws_size; (void)out_size;
    const float* x     = (const float*)d_in[0];
    const float* noise = (const float*)d_in[1];
    const float* W_ih  = (const float*)d_in[2];
    const float* b_ih  = (const float*)d_in[3];
    const float* W_hh  = (const float*)d_in[4];
    const float* b_hh  = (const float*)d_in[5];
    const float* W_out = (const float*)d_in[6];
    const float* b_out = (const float*)d_in[7];
    float* out = (float*)d_out;

    dim3 grid(B_ / BT);        // 64 workgroups
    dim3 block(NTHREADS);      // 8 wave32s
    rnn_fused_kernel<<<grid, block, 0, stream>>>(x, noise, W_ih, b_ih, W_hh, b_hh,
                                                 W_out, b_out, out);
}